// GNLB_13829794693651
// MI455X (gfx1250) — compile-verified
//
#include <hip/hip_runtime.h>
#include <hip/hip_bf16.h>

typedef __attribute__((ext_vector_type(2))) float v2f;
typedef __attribute__((ext_vector_type(8))) float v8f;

#define CDIM 256
#define HW   1024   // 32*32
#define NPIX 8192   // B(8) * HW

// Entire reference network collapses (softmax over size-1 axis == ones) to:
//   rowsum[c] = sum_{c'} wm[c][c']                      (via V_WMMA_F32_16X16X4_F32, B=ones)
//   c3[o]     = bo[o] + sum_c wo[o][c]*(bm[c]+rowsum[c])
//   k[o]      = bl[o] + sum_i wl[o][i]*c3[i]
//   out[b,o,hw] = k[o] + sum_i wl[o][i]*x1_src[b,i,hw]
__global__ __launch_bounds__(256) void gnlb_fused(
    const float* __restrict__ x1_src,
    const float* __restrict__ wm,
    const float* __restrict__ bm,
    const float* __restrict__ wo,
    const float* __restrict__ bo,
    const float* __restrict__ wl,
    const float* __restrict__ bl,
    float* __restrict__ out)
{
    __shared__ float s_rowsum[CDIM];
    __shared__ float s_c3[3];
    __shared__ float s_k[3];

    const int tid  = threadIdx.x;
    const int lane = tid & 31;    // wave32
    const int wave = tid >> 5;    // 8 waves per block

    __builtin_prefetch(wm, 0, 0); // global_prefetch_b8

    // ---- Phase 1: rowsum via WMMA f32 16x16x4, B = all-ones --------------
    // A layout (16x4 f32): lanes 0-15 -> M=lane, VGPR{0,1}=K{0,1};
    //                      lanes 16-31 -> M=lane-16, VGPR{0,1}=K{2,3}.
    const int m     = lane & 15;
    const int khalf = lane >> 4;
    v2f bones = {1.0f, 1.0f};

    for (int t = 0; t < 2; ++t) {
        const int tile = wave * 2 + t;                       // 16 tiles of 16 rows
        const float* arow = wm + (tile * 16 + m) * CDIM + khalf * 2;
        v8f acc = {};
        #pragma unroll 8
        for (int kk = 0; kk < 64; ++kk) {                    // K = 256 in steps of 4
            v2f a;
            a.x = arow[kk * 4 + 0];
            a.y = arow[kk * 4 + 1];
            acc = __builtin_amdgcn_wmma_f32_16x16x4_f32(
                /*neg_a=*/false, a, /*neg_b=*/false, bones,
                /*c_mod=*/(short)0, acc, /*reuse_a=*/false, /*reuse_b=*/false);
        }
        // D layout (16x16 f32): VGPR r, lane 0 -> M=r (N=0); lane 16 -> M=8+r.
        if (lane == 0) {
            #pragma unroll
            for (int r = 0; r < 8; ++r) s_rowsum[tile * 16 + r] = acc[r];
        } else if (lane == 16) {
            #pragma unroll
            for (int r = 0; r < 8; ++r) s_rowsum[tile * 16 + 8 + r] = acc[r];
        }
    }
    __syncthreads();

    // ---- Phase 2: fold the scalar constants ------------------------------
    if (tid < 3) {
        float s = bo[tid];
        const float* wor = wo + tid * CDIM;
        for (int c = 0; c < CDIM; ++c)
            s += wor[c] * (bm[c] + s_rowsum[c]);
        s_c3[tid] = s;
    }
    __syncthreads();
    if (tid < 3) {
        s_k[tid] = bl[tid] + wl[tid * 3 + 0] * s_c3[0]
                           + wl[tid * 3 + 1] * s_c3[1]
                           + wl[tid * 3 + 2] * s_c3[2];
    }
    __syncthreads();

    // ---- Phase 3: per-pixel 3x3 affine transform -------------------------
    const float k0 = s_k[0], k1 = s_k[1], k2 = s_k[2];
    const float w00 = wl[0], w01 = wl[1], w02 = wl[2];
    const float w10 = wl[3], w11 = wl[4], w12 = wl[5];
    const float w20 = wl[6], w21 = wl[7], w22 = wl[8];

    const int pix = blockIdx.x * 256 + tid;   // 0..8191
    const int b   = pix >> 10;
    const int hw  = pix & 1023;
    const float* xp = x1_src + b * (3 * HW) + hw;
    const float x0 = xp[0];
    const float x1 = xp[HW];
    const float x2 = xp[2 * HW];
    float* op = out + b * (3 * HW) + hw;
    op[0]      = k0 + w00 * x0 + w01 * x1 + w02 * x2;
    op[HW]     = k1 + w10 * x0 + w11 * x1 + w12 * x2;
    op[2 * HW] = k2 + w20 * x0 + w21 * x1 + w22 * x2;
}

extern "C" void kernel_launch(void* const* d_in, const int* in_sizes, int n_in,
                              void* d_out, int out_size, void* d_ws, size_t ws_size,
                              hipStream_t stream) {
    // setup_inputs order:
    // 0:x0 1:x1_src 2:w0 3:b0 4:w1 5:b1 6:wm 7:bm 8:wo 9:bo 10:wl 11:bl
    const float* x1_src = (const float*)d_in[1];
    const float* wm     = (const float*)d_in[6];
    const float* bm     = (const float*)d_in[7];
    const float* wo     = (const float*)d_in[8];
    const float* bo     = (const float*)d_in[9];
    const float* wl     = (const float*)d_in[10];
    const float* bl     = (const float*)d_in[11];
    float* out = (float*)d_out;

    gnlb_fused<<<NPIX / 256, 256, 0, stream>>>(x1_src, wm, bm, wo, bo, wl, bl, out);
}